// LayerQuintiles_41601053229441
// MI455X (gfx1250) — compile-verified
//
#include <hip/hip_runtime.h>
#include <stdint.h>

#define SEG   32768
#define NT    512
#define NQ    5
#define LLAY  8

typedef unsigned int u32x4 __attribute__((ext_vector_type(4)));
typedef int          i32x4 __attribute__((ext_vector_type(4)));
typedef int          i32x8 __attribute__((ext_vector_type(8)));

#if defined(__has_builtin)
#if __has_builtin(__builtin_amdgcn_tensor_load_to_lds)
#define HAVE_TDM 1
#endif
#endif

__constant__ float c_q[NQ] = {1.0f, 25.0f, 50.0f, 75.0f, 99.0f};

// map float bits -> unsigned with same total order (no NaNs in input)
__device__ __forceinline__ unsigned mapf(float f) {
  unsigned u = __float_as_uint(f);
  return (u & 0x80000000u) ? ~u : (u ^ 0x80000000u);
}
__device__ __forceinline__ float unmapf(unsigned m) {
  unsigned u = (m & 0x80000000u) ? (m ^ 0x80000000u) : ~m;
  return __uint_as_float(u);
}

__global__ __launch_bounds__(NT) void layer_quintiles_kernel(
    const float* __restrict__ x, float* __restrict__ out) {
  __shared__ float    s_data[SEG];        // 128 KB segment tile
  __shared__ unsigned s_hist[16][256];    // lvl0: per-wave privatized; lvl1-3: rows 0..4 per quantile
  __shared__ float    s_red[NT];
  __shared__ unsigned s_pref[NQ], s_want[NQ], s_cnt[NQ], s_minv[NQ];
  __shared__ int      s_rank[NQ];
  __shared__ float    s_fr[NQ];

  const unsigned tid = threadIdx.x;
  const unsigned seg = blockIdx.x;                      // seg = b*8 + l (contiguous layout)
  const float* __restrict__ src = x + (size_t)seg * SEG;
  const unsigned obase = (seg >> 3) * (LLAY * 7u) + (seg & 7u) * 7u;

  // ---------------- 1. Tensor Data Mover: global -> LDS (async DMA) ----------------
#ifdef HAVE_TDM
  if (tid < 32u) {  // wave 0 only issues the TDM op (EXEC ignored; per-wave issue)
    unsigned lds0 = (unsigned)(uintptr_t)&s_data[0];    // low 32 bits = LDS byte offset
    unsigned long long ga = (unsigned long long)(uintptr_t)src;
    // D# group0: count=1 | lds_addr | global_addr[56:0] | type=2
    u32x4 g0 = {1u, lds0, (unsigned)(ga & 0xFFFFFFFFull),
                (unsigned)((ga >> 32) & 0x01FFFFFFull) | 0x80000000u};
    // D# group1: data_size=4B; tensor_dim0=32768, tensor_dim1=1; tile_dim0=32768,
    // tile_dim1=1; tensor_dim0_stride=32768; tensor_dim1_stride=32768
    i32x8 g1 = {(int)0x00020000u, (int)0x80000000u, (int)0x00010000u, (int)0x80000000u,
                1, 32768, (int)0x80000000u, 0};
    i32x4 g2 = {0, 0, 0, 0};
    i32x4 g3 = {0, 0, 0, 0};
#if defined(__clang_major__) && __clang_major__ >= 23
    i32x8 g4 = {0, 0, 0, 0, 0, 0, 0, 0};
    __builtin_amdgcn_tensor_load_to_lds(g0, g1, g2, g3, g4, 0);
#else
    __builtin_amdgcn_tensor_load_to_lds(g0, g1, g2, g3, 0);
#endif
    __builtin_amdgcn_s_wait_tensorcnt(0);
  }
  __syncthreads();
#else
  for (unsigned i = tid; i < SEG; i += NT) s_data[i] = src[i];
  __syncthreads();
#endif

  // ---------------- 2. mean / var (two-pass, deterministic tree reduce) ----------------
  float ls = 0.0f;
  for (unsigned i = tid; i < SEG; i += NT) ls += s_data[i];
  s_red[tid] = ls;
  __syncthreads();
  for (unsigned s = NT / 2; s > 0; s >>= 1) {
    if (tid < s) s_red[tid] += s_red[tid + s];
    __syncthreads();
  }
  const float mean = s_red[0] * (1.0f / (float)SEG);
  __syncthreads();

  float lv = 0.0f;
  for (unsigned i = tid; i < SEG; i += NT) {
    float d = s_data[i] - mean;
    lv += d * d;
  }
  s_red[tid] = lv;
  __syncthreads();
  for (unsigned s = NT / 2; s > 0; s >>= 1) {
    if (tid < s) s_red[tid] += s_red[tid + s];
    __syncthreads();
  }
  const float var = s_red[0] * (1.0f / (float)SEG);
  if (tid == 0) { out[obase + 0] = mean; out[obase + 1] = var; }

  // ---------------- 3. radix select, all 5 quantiles at once ----------------
  // init: ranks/fracs computed in f32 exactly like jnp: idx = (q/100)*(n-1)
  for (unsigned j = tid; j < 16u * 256u; j += NT) ((unsigned*)s_hist)[j] = 0u;
  if (tid < NQ) {
    float idx = c_q[tid] * 0.01f * 32767.0f;
    int k = (int)idx;
    s_rank[tid] = k;
    s_fr[tid]   = idx - (float)k;
    s_want[tid] = (unsigned)k;
    s_pref[tid] = 0u;
  }
  __syncthreads();

  // level 0 (bits 31:24): one shared scan, per-wave privatized histograms
  {
    const unsigned wrow = tid >> 5;  // wave32 -> wave id 0..15
    for (unsigned i = tid; i < SEG; i += NT) {
      unsigned m = mapf(s_data[i]);
      atomicAdd(&s_hist[wrow][m >> 24], 1u);
    }
    __syncthreads();
    if (tid < 256u) {
      unsigned t = 0u;
      for (int w = 0; w < 16; ++w) t += s_hist[w][tid];
      s_hist[0][tid] = t;
    }
    __syncthreads();
    if (tid < NQ) {
      unsigned w = s_want[tid], cum = 0u, sel = 0u;
      for (int b = 0; b < 256; ++b) {
        unsigned c = s_hist[0][b];
        if (w < cum + c) { sel = (unsigned)b; w -= cum; break; }
        cum += c;
      }
      s_pref[tid] = sel << 24;
      s_want[tid] = w;
    }
    __syncthreads();
  }

  // levels 1..3 (bits 23:16, 15:8, 7:0): 5 per-quantile histograms per scan
  for (int shift = 16; shift >= 0; shift -= 8) {
    for (unsigned j = tid; j < (unsigned)NQ * 256u; j += NT) ((unsigned*)s_hist)[j] = 0u;
    __syncthreads();
    const unsigned mask = 0xFFFFFFFFu << (shift + 8);
    const unsigned p0 = s_pref[0], p1 = s_pref[1], p2 = s_pref[2], p3 = s_pref[3], p4 = s_pref[4];
    for (unsigned i = tid; i < SEG; i += NT) {
      unsigned m = mapf(s_data[i]);
      unsigned mm = m & mask;
      unsigned bin = (m >> shift) & 255u;
      if (mm == p0) atomicAdd(&s_hist[0][bin], 1u);
      if (mm == p1) atomicAdd(&s_hist[1][bin], 1u);
      if (mm == p2) atomicAdd(&s_hist[2][bin], 1u);
      if (mm == p3) atomicAdd(&s_hist[3][bin], 1u);
      if (mm == p4) atomicAdd(&s_hist[4][bin], 1u);
    }
    __syncthreads();
    if (tid < NQ) {
      unsigned w = s_want[tid], cum = 0u, sel = 0u;
      for (int b = 0; b < 256; ++b) {
        unsigned c = s_hist[tid][b];
        if (w < cum + c) { sel = (unsigned)b; w -= cum; break; }
        cum += c;
      }
      s_pref[tid] |= sel << shift;
      s_want[tid] = w;
    }
    __syncthreads();
  }
  // s_pref[q] is now the bit-exact mapped value of sorted[rank_q]

  // ---------------- 4. sorted[rank+1] via count(<=v_k) and min(>v_k) ----------------
  if (tid < NQ) { s_cnt[tid] = 0u; s_minv[tid] = 0xFFFFFFFFu; }
  __syncthreads();
  {
    const unsigned P0 = s_pref[0], P1 = s_pref[1], P2 = s_pref[2], P3 = s_pref[3], P4 = s_pref[4];
    unsigned c0 = 0, c1 = 0, c2 = 0, c3 = 0, c4 = 0;
    unsigned n0 = ~0u, n1 = ~0u, n2 = ~0u, n3 = ~0u, n4 = ~0u;
    for (unsigned i = tid; i < SEG; i += NT) {
      unsigned m = mapf(s_data[i]);
      if (m <= P0) c0++; else if (m < n0) n0 = m;
      if (m <= P1) c1++; else if (m < n1) n1 = m;
      if (m <= P2) c2++; else if (m < n2) n2 = m;
      if (m <= P3) c3++; else if (m < n3) n3 = m;
      if (m <= P4) c4++; else if (m < n4) n4 = m;
    }
    atomicAdd(&s_cnt[0], c0); atomicAdd(&s_cnt[1], c1); atomicAdd(&s_cnt[2], c2);
    atomicAdd(&s_cnt[3], c3); atomicAdd(&s_cnt[4], c4);
    atomicMin(&s_minv[0], n0); atomicMin(&s_minv[1], n1); atomicMin(&s_minv[2], n2);
    atomicMin(&s_minv[3], n3); atomicMin(&s_minv[4], n4);
  }
  __syncthreads();
  if (tid < NQ) {
    float vk = unmapf(s_pref[tid]);
    unsigned kp2 = (unsigned)s_rank[tid] + 2u;
    float vk1 = (s_cnt[tid] >= kp2) ? vk : unmapf(s_minv[tid]);
    out[obase + 2u + tid] = vk + (vk1 - vk) * s_fr[tid];
  }
}

extern "C" void kernel_launch(void* const* d_in, const int* in_sizes, int n_in,
                              void* d_out, int out_size, void* d_ws, size_t ws_size,
                              hipStream_t stream) {
  const float* x = (const float*)d_in[0];  // (B, N) fp32; d_in[1] (pos) encodes identity order
  float* out = (float*)d_out;              // (B, 56) fp32
  int nseg = in_sizes[0] / SEG;            // B * L = 4096
  hipLaunchKernelGGL(layer_quintiles_kernel, dim3(nseg), dim3(NT), 0, stream, x, out);
}